// RWKV_Time_Mixing_63960652972088
// MI455X (gfx1250) — compile-verified
//
#include <hip/hip_runtime.h>

#define N_EMBD 4096

typedef float v2f __attribute__((ext_vector_type(2)));
typedef float v8f __attribute__((ext_vector_type(8)));

// ---------------------------------------------------------------------------
// WMMA fp32 matvec: one 16-row output block per workgroup, K split over 4
// waves (1024 K each), accumulated with V_WMMA_F32_16X16X4_F32.
//
// A-fragment layout (16x4 f32, 2 VGPRs): lanes 0-15 hold row=lane, K={0,1};
// lanes 16-31 hold row=lane-16, K={2,3}.  -> one float2 load per lane.
// B-fragment (4x16): only column N=0 is live: lane 0 holds x[k..k+1],
// lane 16 holds x[k+2..k+3], all other lanes zero.
// D column 0: rows m0..m0+7 in lane 0 (acc[0..7]), rows m0+8..15 in lane 16.
// ---------------------------------------------------------------------------
__device__ __forceinline__ void wmma_matvec_rows16(
    const float* __restrict__ W, const float* __restrict__ x,
    float* __restrict__ y, int m0)
{
    const int tid  = threadIdx.x;
    const int lane = tid & 31;
    const int wv   = tid >> 5;          // 0..3 (4 waves / block)
    const int half = lane >> 4;         // 0 or 1
    const int mrow = lane & 15;

    const float* Wp = W + (size_t)(m0 + mrow) * N_EMBD + (half << 1);
    const float* xp = x + (half << 1);
    const int kbase = wv * (N_EMBD / 4);          // 1024 K per wave
    const float msel = (mrow == 0) ? 1.0f : 0.0f; // keep only B column 0

    v8f acc = {0.f, 0.f, 0.f, 0.f, 0.f, 0.f, 0.f, 0.f};

    // 8 WMMA-chunks (32 K) per iteration: 8 b64 weight loads in flight/wave.
    for (int k = kbase; k < kbase + (N_EMBD / 4); k += 32) {
        v2f a[8], b[8];
#pragma unroll
        for (int u = 0; u < 8; ++u)
            a[u] = *(const v2f*)(Wp + k + 4 * u);
#pragma unroll
        for (int u = 0; u < 8; ++u)
            b[u] = *(const v2f*)(xp + k + 4 * u) * msel;
#pragma unroll
        for (int u = 0; u < 8; ++u)
            acc = __builtin_amdgcn_wmma_f32_16x16x4_f32(
                false, a[u], false, b[u], (short)0, acc, false, false);
    }

    // cross-wave reduction of the 4 K-partials
    __shared__ float red[4][16];
    if (lane == 0) {
#pragma unroll
        for (int i = 0; i < 8; ++i) red[wv][i] = acc[i];
    } else if (lane == 16) {
#pragma unroll
        for (int i = 0; i < 8; ++i) red[wv][8 + i] = acc[i];
    }
    __syncthreads();
    if (tid < 16)
        y[m0 + tid] = red[0][tid] + red[1][tid] + red[2][tid] + red[3][tid];
}

__global__ void rwkv_matvec3_kernel(const float* __restrict__ kw,
                                    const float* __restrict__ vw,
                                    const float* __restrict__ rw,
                                    const float* __restrict__ xin,  // xk,xv,xr
                                    float* __restrict__ yout)       // kk,vv,rr
{
    const int which = blockIdx.y;
    const float* W = (which == 0) ? kw : (which == 1) ? vw : rw;
    const float* x = xin + (size_t)which * N_EMBD;
    float* y       = yout + (size_t)which * N_EMBD;
    wmma_matvec_rows16(W, x, y, blockIdx.x * 16);
}

__global__ void rwkv_matvec1_kernel(const float* __restrict__ W,
                                    const float* __restrict__ x,
                                    float* __restrict__ y)
{
    wmma_matvec_rows16(W, x, y, blockIdx.x * 16);
}

// ---------------------------------------------------------------------------
// LayerNorm + time-shift mixing.  1 block, 256 threads, 16 elems/thread.
// Writes x into new_state row 5i+1 and xk/xv/xr into workspace.
// ---------------------------------------------------------------------------
__global__ void rwkv_ln_mix_kernel(const float* __restrict__ inp,
                                   const float* __restrict__ state,
                                   const float* __restrict__ tmk,
                                   const float* __restrict__ tmv,
                                   const float* __restrict__ tmr,
                                   const float* __restrict__ lnw,
                                   const float* __restrict__ lnb,
                                   const int* __restrict__ layer,
                                   float* __restrict__ new_state,
                                   float* __restrict__ ws)
{
    __shared__ float ssum[8], ssq[8];
    const int tid = threadIdx.x;

    float v[16];
    float s = 0.f, q = 0.f;
#pragma unroll
    for (int i = 0; i < 16; ++i) {
        float t = inp[tid + i * 256];
        v[i] = t; s += t; q += t * t;
    }
#pragma unroll
    for (int off = 16; off > 0; off >>= 1) {
        s += __shfl_xor(s, off, 32);
        q += __shfl_xor(q, off, 32);
    }
    if ((tid & 31) == 0) { ssum[tid >> 5] = s; ssq[tid >> 5] = q; }
    __syncthreads();
    float S = 0.f, Q = 0.f;
#pragma unroll
    for (int i = 0; i < 8; ++i) { S += ssum[i]; Q += ssq[i]; }

    const float mu  = S * (1.0f / N_EMBD);
    const float var = Q * (1.0f / N_EMBD) - mu * mu;
    const float inv = rsqrtf(var + 1e-5f);

    const int li = *layer;
    const float* prev = state + (size_t)(5 * li + 1) * N_EMBD;
    float* xrow = new_state + (size_t)(5 * li + 1) * N_EMBD;

#pragma unroll
    for (int i = 0; i < 16; ++i) {
        const int j = tid + i * 256;
        const float x = (v[i] - mu) * inv * lnw[j] + lnb[j];
        const float p = prev[j];
        xrow[j]               = x;
        ws[j]                 = x * tmk[j] + p * (1.0f - tmk[j]);
        ws[N_EMBD + j]        = x * tmv[j] + p * (1.0f - tmv[j]);
        ws[2 * N_EMBD + j]    = x * tmr[j] + p * (1.0f - tmr[j]);
    }
}

// ---------------------------------------------------------------------------
// WKV recurrence + state update + r*wkv (input of the ow matvec).
// ---------------------------------------------------------------------------
__global__ void rwkv_wkv_kernel(const float* __restrict__ ws,     // kk,vv,rr at 3..5*N
                                const float* __restrict__ state,
                                const float* __restrict__ time_first,
                                const float* __restrict__ time_decay,
                                const int* __restrict__ layer,
                                float* __restrict__ new_state,
                                float* __restrict__ rwkv)
{
    const int j  = blockIdx.x * blockDim.x + threadIdx.x;
    const int li = *layer;

    const float kk = ws[3 * N_EMBD + j];
    const float vv = ws[4 * N_EMBD + j];
    const float rr = ws[5 * N_EMBD + j];
    const float r  = 1.0f / (1.0f + expf(-rr));

    const float aa = state[(size_t)(5 * li + 2) * N_EMBD + j];
    const float bb = state[(size_t)(5 * li + 3) * N_EMBD + j];
    const float pp = state[(size_t)(5 * li + 4) * N_EMBD + j];

    const float ww = time_first[j] + kk;
    const float p  = fmaxf(pp, ww);
    const float e1 = expf(pp - p);
    const float e2 = expf(ww - p);
    const float a  = e1 * aa + e2 * vv;
    const float b  = e1 * bb + e2;

    const float ww2 = pp + time_decay[j];
    const float p2  = fmaxf(ww2, kk);
    const float e1s = expf(ww2 - p2);
    const float e2s = expf(kk - p2);

    new_state[(size_t)(5 * li + 2) * N_EMBD + j] = e1s * aa + e2s * vv;
    new_state[(size_t)(5 * li + 3) * N_EMBD + j] = e1s * bb + e2s;
    new_state[(size_t)(5 * li + 4) * N_EMBD + j] = p2;

    rwkv[j] = r * (a / b);
}

// ---------------------------------------------------------------------------
extern "C" void kernel_launch(void* const* d_in, const int* in_sizes, int n_in,
                              void* d_out, int out_size, void* d_ws, size_t ws_size,
                              hipStream_t stream)
{
    (void)n_in; (void)out_size; (void)ws_size;

    const float* input      = (const float*)d_in[0];
    const float* state      = (const float*)d_in[1];
    const float* time_mix_k = (const float*)d_in[2];
    const float* time_mix_v = (const float*)d_in[3];
    const float* time_mix_r = (const float*)d_in[4];
    const float* time_first = (const float*)d_in[5];
    const float* time_decay = (const float*)d_in[6];
    const float* kw         = (const float*)d_in[7];
    const float* vw         = (const float*)d_in[8];
    const float* rw         = (const float*)d_in[9];
    const float* ow         = (const float*)d_in[10];
    const float* ln_weight  = (const float*)d_in[11];
    const float* ln_bias    = (const float*)d_in[12];
    const int*   layer      = (const int*)d_in[13];

    const size_t state_elems = (size_t)in_sizes[1];     // 5*32*4096
    float* new_state = (float*)d_out;
    float* out_vec   = (float*)d_out + state_elems;

    float* ws   = (float*)d_ws;      // [0..3N): xk,xv,xr  [3N..6N): kk,vv,rr
    float* rwkv = ws + 6 * N_EMBD;   // [6N..7N): r * wkv

    // new_state starts as a copy of state; kernels overwrite 4 rows.
    hipMemcpyAsync(new_state, state, state_elems * sizeof(float),
                   hipMemcpyDeviceToDevice, stream);

    rwkv_ln_mix_kernel<<<1, 256, 0, stream>>>(
        input, state, time_mix_k, time_mix_v, time_mix_r,
        ln_weight, ln_bias, layer, new_state, ws);

    dim3 grid3(N_EMBD / 16, 3);
    rwkv_matvec3_kernel<<<grid3, 128, 0, stream>>>(
        kw, vw, rw, ws, ws + 3 * N_EMBD);

    rwkv_wkv_kernel<<<N_EMBD / 256, 256, 0, stream>>>(
        ws, state, time_first, time_decay, layer, new_state, rwkv);

    rwkv_matvec1_kernel<<<N_EMBD / 16, 128, 0, stream>>>(ow, rwkv, out_vec);
}